// QuanvLayer_40931038331588
// MI455X (gfx1250) — compile-verified
//
#include <hip/hip_runtime.h>
#include <math.h>

// QuanvLayer collapsed analytically (L=1 BasicEntangler, 2x2 window, C=2):
//   per pixel: alpha = norm(ch0), beta = norm(ch1)
//   z_i(pixel) = cos(phi_i)*cos(beta)*cos(alpha) - sin(phi_i)*sin(alpha)
//   out[q=0]=z1*z2*z3, out[1]=z0*z1, out[2]=z0*z1*z2, out[3]=z0*z1*z2*z3
// (z_i taken from the 4 pixels of the window; phi_i = weights[0,i])

#define C_IN   2
#define HH     224
#define WW     224
#define OH     223
#define OW     223
#define TILE_W 32
#define TILE_H 8
#define PW     (TILE_W + 1)   // 33 pixels wide per tile
#define PH     (TILE_H + 1)   // 9 pixels tall per tile
#define PWP    36             // padded LDS stride: 36*4 = 144 B (16B aligned rows)
#define NPIX   (PH * PW)      // 297
#define PLANE  (HH * WW)      // 50176
#define SAMP   (C_IN * PLANE) // 100352
#define PI_F   3.14159265358979323846f

// CDNA5 async global->LDS copies (VGLOBAL ops 96/98, tracked by ASYNCcnt).
__device__ __forceinline__ void async_g2lds_b32(void* ldsPtr, const void* gptr) {
  unsigned int laddr = (unsigned int)(unsigned long long)ldsPtr; // low 32 bits = LDS offset
  asm volatile("global_load_async_to_lds_b32 %0, %1, off"
               :: "v"(laddr), "v"(gptr)
               : "memory");
}
__device__ __forceinline__ void async_g2lds_b128(void* ldsPtr, const void* gptr) {
  unsigned int laddr = (unsigned int)(unsigned long long)ldsPtr;
  asm volatile("global_load_async_to_lds_b128 %0, %1, off"
               :: "v"(laddr), "v"(gptr)
               : "memory");
}
__device__ __forceinline__ void wait_asynccnt0() {
  asm volatile("s_wait_asynccnt 0" ::: "memory");
}

// ws layout (floats): [0..B-1]=pmin, [B..2B-1]=scale(pi/range), [2B..2B+3]=cos(w), [2B+4..2B+7]=sin(w)
__global__ __launch_bounds__(256)
void quanv_stats(const float* __restrict__ x, const float* __restrict__ w,
                 float* __restrict__ ws, int B) {
  __shared__ float smin[256];
  __shared__ float smax[256];
  const int b = blockIdx.x, tid = threadIdx.x;
  const float4* p4 = (const float4*)(x + (size_t)b * SAMP); // SAMP % 4 == 0, 16B aligned
  float m = 3.4028235e38f, M = -3.4028235e38f;
  for (int i = tid; i < SAMP / 4; i += 256) {               // 98 iterations, b128 loads
    float4 v = p4[i];
    m = fminf(m, fminf(fminf(v.x, v.y), fminf(v.z, v.w)));
    M = fmaxf(M, fmaxf(fmaxf(v.x, v.y), fmaxf(v.z, v.w)));
  }
  smin[tid] = m; smax[tid] = M;
  __syncthreads();
  for (int s = 128; s > 0; s >>= 1) {
    if (tid < s) {
      smin[tid] = fminf(smin[tid], smin[tid + s]);
      smax[tid] = fmaxf(smax[tid], smax[tid + s]);
    }
    __syncthreads();
  }
  if (tid == 0) {
    ws[b]     = smin[0];
    ws[B + b] = PI_F / ((smax[0] - smin[0]) + 1e-8f);
  }
  if (b == 0 && tid < 4) {
    float ang = w[tid];          // weights in [0, 2*pi)
    ws[2 * B + tid]     = cosf(ang);
    ws[2 * B + 4 + tid] = sinf(ang);
  }
}

__global__ __launch_bounds__(256)
void quanv_main(const float* __restrict__ x, const float* __restrict__ ws,
                float* __restrict__ out, int B) {
  // ta: raw c0 then t0 = cos(beta)*cos(alpha);  tb: raw c1 then t1 = sin(alpha)
  __shared__ __align__(16) float ta[PH][PWP];
  __shared__ __align__(16) float tb[PH][PWP];
  const int tid = threadIdx.x;
  const int b   = blockIdx.z;
  const int x0  = blockIdx.x * TILE_W;
  const int y0  = blockIdx.y * TILE_H;
  const float* pa = x + (size_t)b * SAMP;  // channel-0 plane
  const float* pb = pa + PLANE;            // channel-1 plane

  // Phase 1: async-stage the (PH x PW) pixel tile (both channels) into LDS.
  // 18 channel-rows, each = 8x async b128 (j=0..31, always in-bounds since
  // x0+31 <= 223) + 1x async b32 tail (j=32, clamped). Bottom row clamps gh.
  {
    const int r = tid / 9;            // channel-row id 0..17
    const int o = tid - r * 9;        // op id 0..8 within the row
    if (r < 2 * PH) {
      const int i  = r >> 1;
      const int ch = r & 1;
      int gh = y0 + i; if (gh > HH - 1) gh = HH - 1;
      const float* src = (ch ? pb : pa) + gh * WW;
      float*       dst = ch ? &tb[i][0] : &ta[i][0];
      if (o < 8) {
        async_g2lds_b128(dst + 4 * o, src + x0 + 4 * o);   // 16B-aligned both sides
      } else {
        int gw = x0 + TILE_W; if (gw > WW - 1) gw = WW - 1;
        async_g2lds_b32(dst + TILE_W, src + gw);
      }
    }
  }
  wait_asynccnt0();   // my async loads have landed in LDS
  __syncthreads();    // everyone's have

  const float pmin  = ws[b];
  const float scale = ws[B + b];

  // Phase 2: per-pixel trig, computed once, reused by 4 output patches.
  // In-place overwrite is safe: each LDS slot is read+written by one thread only.
  for (int k = tid; k < NPIX; k += 256) {
    int i = k / PW, j = k - i * PW;
    float a  = (ta[i][j] - pmin) * scale;   // alpha in [0, pi]
    float be = (tb[i][j] - pmin) * scale;   // beta  in [0, pi]
    float ca = __cosf(a), sa = __sinf(a), cb = __cosf(be);
    ta[i][j] = cb * ca;
    tb[i][j] = sa;
  }
  __syncthreads();

  const float cw0 = ws[2 * B + 0], cw1 = ws[2 * B + 1];
  const float cw2 = ws[2 * B + 2], cw3 = ws[2 * B + 3];
  const float sw0 = ws[2 * B + 4], sw1 = ws[2 * B + 5];
  const float sw2 = ws[2 * B + 6], sw3 = ws[2 * B + 7];

  // Phase 3: one output patch per thread (lanes sweep j -> conflict-free LDS,
  // coalesced stores into each of the 4 q-planes).
  const int j  = tid & (TILE_W - 1);
  const int i  = tid >> 5;
  const int oh = y0 + i, ow = x0 + j;
  if (oh < OH && ow < OW) {
    float z0 = fmaf(cw0, ta[i    ][j    ], -sw0 * tb[i    ][j    ]);
    float z1 = fmaf(cw1, ta[i    ][j + 1], -sw1 * tb[i    ][j + 1]);
    float z2 = fmaf(cw2, ta[i + 1][j    ], -sw2 * tb[i + 1][j    ]);
    float z3 = fmaf(cw3, ta[i + 1][j + 1], -sw3 * tb[i + 1][j + 1]);
    float p01 = z0 * z1, p23 = z2 * z3;
    const size_t Q = (size_t)OH * OW;
    size_t o = (size_t)b * 4 * Q + (size_t)oh * OW + ow;
    out[o]         = z1 * p23;   // q0 = z1*z2*z3
    out[o + Q]     = p01;        // q1 = z0*z1
    out[o + 2 * Q] = p01 * z2;   // q2 = z0*z1*z2
    out[o + 3 * Q] = p01 * p23;  // q3 = z0*z1*z2*z3
  }
}

extern "C" void kernel_launch(void* const* d_in, const int* in_sizes, int n_in,
                              void* d_out, int out_size, void* d_ws, size_t ws_size,
                              hipStream_t stream) {
  (void)n_in; (void)out_size; (void)ws_size;
  const float* x = (const float*)d_in[0];
  const float* w = (const float*)d_in[1];   // [1,4] weights (L=1)
  float* out = (float*)d_out;
  float* ws  = (float*)d_ws;
  const int B = in_sizes[0] / SAMP;         // 8 for the reference shapes

  quanv_stats<<<dim3(B), dim3(256), 0, stream>>>(x, w, ws, B);

  dim3 grid((OW + TILE_W - 1) / TILE_W,     // 7
            (OH + TILE_H - 1) / TILE_H,     // 28
            B);
  quanv_main<<<grid, dim3(256), 0, stream>>>(x, ws, out, B);
}